// SparsePrePrunedLayer_35510789604085
// MI455X (gfx1250) — compile-verified
//
#include <hip/hip_runtime.h>
#include <stdint.h>

// ---------------------------------------------------------------------------
// COO SpMV  y = W*x + bias   (rows sorted, avg 64 nnz/row)
// Memory-bound: ~154 MB streamed -> ~6.6 us floor @ 23.3 TB/s. FLOPs trivial.
// CDNA5 path: async global->LDS double-buffered streaming (ASYNCcnt),
// wave32 shuffle segmented reduction, one f32 global atomic per row-run.
// ---------------------------------------------------------------------------

#define WAVE            32
#define ELEMS_PER_LANE  4
#define CHUNK           (WAVE * ELEMS_PER_LANE)   // 128 nnz per wave-chunk
#define WAVES_PER_BLOCK 8
#define BLOCK           (WAVE * WAVES_PER_BLOCK)  // 256 threads
#define BUF_BYTES       1536                      // 512B vals + 512B cols + 512B rows
#define SLICE_BYTES     (2 * BUF_BYTES)           // double buffer per wave

// Matches the builtin's parameter pointee type exactly (from the diagnostic):
// 'int __attribute__((vector_size(4*sizeof(int))))'
typedef int v4i_t __attribute__((vector_size(16)));

typedef __attribute__((address_space(3))) void as3_void_t;

__device__ __forceinline__ void async_g2l_b128(const void* gsrc, void* ldst) {
#if __has_builtin(__builtin_amdgcn_global_load_async_to_lds_b128)
  // Generic pointers implicitly convert to the __device__/__shared__
  // address-space-qualified parameters in HIP.
  __builtin_amdgcn_global_load_async_to_lds_b128(
      (v4i_t*)(void*)gsrc, (v4i_t*)ldst, 0, 0);
#else
  unsigned loff = (unsigned)(uintptr_t)(as3_void_t*)ldst;
  asm volatile("global_load_async_to_lds_b128 %0, %1, off"
               :: "v"(loff), "v"(gsrc) : "memory");
#endif
}

__device__ __forceinline__ void wait_async_le3() {
#if __has_builtin(__builtin_amdgcn_s_wait_asynccnt)
  __builtin_amdgcn_s_wait_asynccnt(3);
#else
  asm volatile("s_wait_asynccnt 0x3" ::: "memory");
#endif
}

__device__ __forceinline__ void wait_async_le0() {
#if __has_builtin(__builtin_amdgcn_s_wait_asynccnt)
  __builtin_amdgcn_s_wait_asynccnt(0);
#else
  asm volatile("s_wait_asynccnt 0x0" ::: "memory");
#endif
}

__device__ __forceinline__ void atomic_add_f32(float* p, float v) {
  __hip_atomic_fetch_add(p, v, __ATOMIC_RELAXED, __HIP_MEMORY_SCOPE_AGENT);
}

// y[i] = bias[i]  (d_out is poisoned before timing; must fully rewrite)
__global__ void spmv_init_kernel(const float* __restrict__ bias,
                                 float* __restrict__ y, int n) {
  int i = blockIdx.x * blockDim.x + threadIdx.x;
  if (i < n) y[i] = bias[i];
}

__global__ __launch_bounds__(BLOCK)
void spmv_coo_kernel(const float* __restrict__ vals,
                     const float* __restrict__ x,
                     const int*   __restrict__ rows,
                     const int*   __restrict__ cols,
                     float*       __restrict__ y,
                     int nnz, int nchunks) {
  __shared__ __align__(16) unsigned char smem[WAVES_PER_BLOCK * SLICE_BYTES];

  const int lane      = threadIdx.x & (WAVE - 1);
  const int waveInBlk = threadIdx.x >> 5;
  const int gwave     = blockIdx.x * WAVES_PER_BLOCK + waveInBlk;
  const int nwaves    = gridDim.x * WAVES_PER_BLOCK;

  unsigned char* slice = smem + waveInBlk * SLICE_BYTES;

  int cur    = gwave;
  int bufsel = 0;

  // Prologue: kick off async staging of the first full chunk.
  if (cur < nchunks) {
    int base = cur * CHUNK;
    if (base + CHUNK <= nnz) {
      unsigned char* buf = slice;
      async_g2l_b128(vals + base + lane * 4, buf +        lane * 16);
      async_g2l_b128(cols + base + lane * 4, buf +  512 + lane * 16);
      async_g2l_b128(rows + base + lane * 4, buf + 1024 + lane * 16);
    }
  }

  while (cur < nchunks) {
    const int nxt = cur + nwaves;
    bool nxtIssued = false;
    if (nxt < nchunks) {
      int nbase = nxt * CHUNK;
      if (nbase + CHUNK <= nnz) {   // wave-uniform; EXEC all-ones at issue
        nxtIssued = true;
        unsigned char* nbuf = slice + (bufsel ^ 1) * BUF_BYTES;
        async_g2l_b128(vals + nbase + lane * 4, nbuf +        lane * 16);
        async_g2l_b128(cols + nbase + lane * 4, nbuf +  512 + lane * 16);
        async_g2l_b128(rows + nbase + lane * 4, nbuf + 1024 + lane * 16);
      }
    }

    const int base = cur * CHUNK;
    if (base + CHUNK <= nnz) {
      // Async loads complete in order: <=3 pending means our 3 are done.
      if (nxtIssued) wait_async_le3(); else wait_async_le0();

      unsigned char* buf = slice + bufsel * BUF_BYTES;
      float4 v4 = ((const float4*)(buf       ))[lane];   // ds_load_b128
      int4   c4 = ((const int4*)  (buf +  512))[lane];
      int4   r4 = ((const int4*)  (buf + 1024))[lane];

      // gather x (L2-resident, 800 KB)
      float p0 = v4.x * x[c4.x];
      float p1 = v4.y * x[c4.y];
      float p2 = v4.z * x[c4.z];
      float p3 = v4.w * x[c4.w];

      // --- within-lane run merge (4 contiguous elements) ---
      const int headRow = r4.x;
      float headVal = 0.0f;
      float acc = p0;
      int   curRow = r4.x;
      bool  broke = false;

      if (r4.y == curRow) { acc += p1; }
      else { if (!broke) { headVal = acc; broke = true; }
             else atomic_add_f32(&y[curRow], acc);
             curRow = r4.y; acc = p1; }
      if (r4.z == curRow) { acc += p2; }
      else { if (!broke) { headVal = acc; broke = true; }
             else atomic_add_f32(&y[curRow], acc);
             curRow = r4.z; acc = p2; }
      if (r4.w == curRow) { acc += p3; }
      else { if (!broke) { headVal = acc; broke = true; }
             else atomic_add_f32(&y[curRow], acc);
             curRow = r4.w; acc = p3; }

      const int   tailRow = curRow;
      const float tailVal = acc;
      const int   uniform = broke ? 0 : 1;  // lane is a single run

      // --- cross-lane carry chain (log-step segmented scan, wave32) ---
      int   prevTailRow = __shfl_up(tailRow, 1, WAVE);
      float prevTailVal = __shfl_up(tailVal, 1, WAVE);
      int   prevUniform = __shfl_up(uniform, 1, WAVE);

      int   link = (lane > 0) && (prevTailRow == headRow);
      float c    = link ? prevTailVal : 0.0f;
      int   open = link && prevUniform;  // chain may extend further back

      #pragma unroll
      for (int s = 1; s <= 16; s <<= 1) {
        float pc = __shfl_up(c,    s, WAVE);
        int   po = __shfl_up(open, s, WAVE);
        c    += open ? pc : 0.0f;
        open &= po;
      }
      const float carry = c;

      int nextHeadRow = __shfl_down(headRow, 1, WAVE);
      const bool tailCont = (lane < WAVE - 1) && (nextHeadRow == tailRow);

      // One atomic per maximal row-run intersecting this chunk (~3/chunk).
      if (!uniform)
        atomic_add_f32(&y[headRow], headVal + carry);
      if (!tailCont)
        atomic_add_f32(&y[tailRow], tailVal + (uniform ? carry : 0.0f));
    } else {
      // Tail chunk (rare / possibly absent): guarded direct path.
      #pragma unroll
      for (int j = 0; j < ELEMS_PER_LANE; ++j) {
        int idx = base + lane * ELEMS_PER_LANE + j;
        if (idx < nnz) {
          float p = vals[idx] * x[cols[idx]];
          atomic_add_f32(&y[rows[idx]], p);
        }
      }
    }

    cur = nxt;
    bufsel ^= 1;
  }
}

extern "C" void kernel_launch(void* const* d_in, const int* in_sizes, int n_in,
                              void* d_out, int out_size, void* d_ws, size_t ws_size,
                              hipStream_t stream) {
  // setup_inputs order: vals, x, bias, rows, cols, n_rows
  const float* vals = (const float*)d_in[0];
  const float* x    = (const float*)d_in[1];
  const float* bias = (const float*)d_in[2];
  const int*   rows = (const int*)d_in[3];
  const int*   cols = (const int*)d_in[4];

  const int nnz    = in_sizes[0];
  const int n_rows = in_sizes[2];   // host-side, avoids reading device scalar
  float* y = (float*)d_out;

  spmv_init_kernel<<<(n_rows + 255) / 256, 256, 0, stream>>>(bias, y, n_rows);

  const int nchunks = (nnz + CHUNK - 1) / CHUNK;
  int blocks = (nchunks + WAVES_PER_BLOCK - 1) / WAVES_PER_BLOCK;
  if (blocks > 2048) blocks = 2048;   // grid-stride over chunks
  if (blocks < 1) blocks = 1;

  spmv_coo_kernel<<<blocks, BLOCK, 0, stream>>>(vals, x, rows, cols, y,
                                                nnz, nchunks);
}